// RelGraphConvLayer_10763188044022
// MI455X (gfx1250) — compile-verified
//
#include <hip/hip_runtime.h>
#include <hip/hip_bf16.h>

typedef float v2f __attribute__((ext_vector_type(2)));
typedef float v8f __attribute__((ext_vector_type(8)));

#define N_AUTHOR 300000
#define N_INST   8000
#define N_PAPER  500000
#define F        128

// ---- workspace layout (units: floats) ----
#define SZ_PAPER   ((size_t)N_PAPER * F)    // 64,000,000
#define SZ_AUTHOR  ((size_t)N_AUTHOR * F)   // 38,400,000
#define SZ_INST    ((size_t)N_INST * F)     //  1,024,000
#define OFF_AGG_WRITES  ((size_t)0)
#define OFF_AGG_CITES   (OFF_AGG_WRITES + SZ_PAPER)            //  64,000,000
#define OFF_AGG_REV     (OFF_AGG_CITES  + SZ_PAPER)            // 128,000,000
#define OFF_AGG_AFFIL   (OFF_AGG_REV    + SZ_AUTHOR)           // 166,400,000
#define OFF_DEG_WRITES  (OFF_AGG_AFFIL  + SZ_INST)             // 167,424,000
#define OFF_DEG_CITES   (OFF_DEG_WRITES + (size_t)N_PAPER)     // 167,924,000
#define OFF_DEG_REV     (OFF_DEG_CITES  + (size_t)N_PAPER)     // 168,424,000
#define OFF_DEG_AFFIL   (OFF_DEG_REV    + (size_t)N_AUTHOR)    // 168,724,000
#define OFF_WEIGHTS     (OFF_DEG_AFFIL + (size_t)N_INST)       // 168,732,000
#define ZERO_FLOATS     OFF_WEIGHTS                            // zero aggs + degs

// ------------------------------------------------------------------
// Phase 0: zero the accumulators / degree arrays (float4 stores)
// ------------------------------------------------------------------
__global__ void zero_kernel(float4* __restrict__ p, long n4) {
  long i = (long)blockIdx.x * blockDim.x + threadIdx.x;
  if (i < n4) p[i] = make_float4(0.f, 0.f, 0.f, 0.f);
}

// ------------------------------------------------------------------
// Phase 1: weight[r] = sum_b w_comp[r,b] * basis_w[b]   (4 x 128 x 128)
// ------------------------------------------------------------------
__global__ void make_weights(const float* __restrict__ basis,
                             const float* __restrict__ comp,
                             float* __restrict__ wout) {
  int idx = blockIdx.x * blockDim.x + threadIdx.x;
  if (idx >= 4 * F * F) return;
  int r = idx >> 14;          // / 16384
  int e = idx & (F * F - 1);
  wout[idx] = comp[r * 2 + 0] * basis[e] + comp[r * 2 + 1] * basis[F * F + e];
}

// ------------------------------------------------------------------
// Phase 2: per-edge scatter of RAW src features (agg-first) + degree.
// One wave per edge: coalesced 512B gather, 128 fp32 atomics.
// ------------------------------------------------------------------
__global__ __launch_bounds__(256)
void edge_scatter(const float* __restrict__ x,
                  const int* __restrict__ src,
                  const int* __restrict__ dst,
                  float* __restrict__ agg,
                  float* __restrict__ deg,
                  int nE) {
  int gid  = blockIdx.x * blockDim.x + threadIdx.x;
  int e    = gid >> 5;
  int lane = threadIdx.x & 31;
  if (e >= nE) return;
  int s = src[e];
  int d = dst[e];
  float4 v = ((const float4*)(x + (size_t)s * F))[lane];
  float* a = agg + (size_t)d * F + lane * 4;
  unsafeAtomicAdd(a + 0, v.x);
  unsafeAtomicAdd(a + 1, v.y);
  unsafeAtomicAdd(a + 2, v.z);
  unsafeAtomicAdd(a + 3, v.w);
  if (lane == 0) unsafeAtomicAdd(deg + d, 1.0f);
}

// ------------------------------------------------------------------
// Phase 3: WMMA f32 GEMM + finalize.
// Each wave computes a 32(M) x 128(N) strip (2 M-tiles x 8 N-tiles):
//   out = relu( sum_t scale_t(row) * (A_t @ W_t) + bias )
// Each B fragment is reused by 2 WMMAs (the two M-tiles), halving
// VMEM traffic per matrix op. scale_t = 1/max(deg_t,1) folded into the
// A fragment (A-fragment lane == M row); self-loop term deg == nullptr.
// A 16x4 f32 fragment: lane&15 = M row, (lane>>4)*2 = K offset.
// B 4x16  fragment: lane&15 = N col, (lane>>4)*2 = K offset.
// C/D: vgpr v, halfwave g -> M = v + 8g, N = lane&15.
// ------------------------------------------------------------------
__global__ __launch_bounds__(256)
void gemm_finalize(int n_strips,   // rows / 32
                   const float* __restrict__ A0, const float* __restrict__ W0, const float* __restrict__ D0,
                   const float* __restrict__ A1, const float* __restrict__ W1, const float* __restrict__ D1,
                   const float* __restrict__ A2, const float* __restrict__ W2, const float* __restrict__ D2,
                   const float* __restrict__ bias, float* __restrict__ out) {
  int lane   = threadIdx.x & 31;
  int waveid = threadIdx.x >> 5;
  int strip  = blockIdx.x * (blockDim.x >> 5) + waveid;
  if (strip >= n_strips) return;              // wave-uniform: EXEC stays all-ones

  int l16  = lane & 15;
  int koff = (lane >> 4) * 2;                 // 0 or 2
  int row0 = strip * 32 + l16;                // M-tile 0 row for this lane
  int row1 = row0 + 16;                       // M-tile 1 row for this lane

  v8f acc[2][8];
#pragma unroll
  for (int mt = 0; mt < 2; ++mt)
#pragma unroll
    for (int i = 0; i < 8; ++i)
      acc[mt][i] = (v8f){0.f, 0.f, 0.f, 0.f, 0.f, 0.f, 0.f, 0.f};

  const float* As[3] = {A0, A1, A2};
  const float* Ws[3] = {W0, W1, W2};
  const float* Ds[3] = {D0, D1, D2};

  for (int t = 0; t < 3; ++t) {
    const float* A = As[t];
    if (A == nullptr) continue;
    const float* W = Ws[t];
    const float* D = Ds[t];
    float s0 = 1.0f, s1 = 1.0f;
    if (D != nullptr) {
      s0 = 1.0f / fmaxf(D[row0], 1.0f);
      s1 = 1.0f / fmaxf(D[row1], 1.0f);
    }
    const float* ar0 = A + (size_t)row0 * F;
    const float* ar1 = A + (size_t)row1 * F;
    for (int k = 0; k < F; k += 4) {
      v2f a0 = *(const v2f*)(ar0 + k + koff);   // A[row0, k+koff .. +1]
      a0.x *= s0; a0.y *= s0;
      v2f a1 = *(const v2f*)(ar1 + k + koff);   // A[row1, k+koff .. +1]
      a1.x *= s1; a1.y *= s1;
      const float* w0 = W + (size_t)(k + koff) * F;
      const float* w1 = w0 + F;
#pragma unroll
      for (int nt = 0; nt < 8; ++nt) {
        v2f b;
        b.x = w0[nt * 16 + l16];                // W[k+koff  , n]
        b.y = w1[nt * 16 + l16];                // W[k+koff+1, n]
        acc[0][nt] = __builtin_amdgcn_wmma_f32_16x16x4_f32(
            false, a0, false, b, (short)0, acc[0][nt], false, false);
        acc[1][nt] = __builtin_amdgcn_wmma_f32_16x16x4_f32(
            false, a1, false, b, (short)0, acc[1][nt], false, false);
      }
    }
  }

  // bias + relu + store
#pragma unroll
  for (int mt = 0; mt < 2; ++mt) {
#pragma unroll
    for (int nt = 0; nt < 8; ++nt) {
      int n = nt * 16 + l16;
      float bv = bias[n];
#pragma unroll
      for (int v = 0; v < 8; ++v) {
        int m = strip * 32 + mt * 16 + v + 8 * (lane >> 4);
        float val = acc[mt][nt][v] + bv;
        out[(size_t)m * F + n] = fmaxf(val, 0.0f);
      }
    }
  }
}

// ------------------------------------------------------------------
extern "C" void kernel_launch(void* const* d_in, const int* in_sizes, int n_in,
                              void* d_out, int out_size, void* d_ws, size_t ws_size,
                              hipStream_t stream) {
  const float* x_author = (const float*)d_in[0];
  const float* x_inst   = (const float*)d_in[1];
  const float* x_paper  = (const float*)d_in[2];
  const float* basis_w  = (const float*)d_in[3];
  const float* w_comp   = (const float*)d_in[4];
  const float* self_w   = (const float*)d_in[5];
  const float* bias     = (const float*)d_in[6];
  const int* writes_src = (const int*)d_in[7];
  const int* writes_dst = (const int*)d_in[8];
  const int* cites_src  = (const int*)d_in[9];
  const int* cites_dst  = (const int*)d_in[10];
  const int* rev_src    = (const int*)d_in[11];
  const int* rev_dst    = (const int*)d_in[12];
  const int* affil_src  = (const int*)d_in[13];
  const int* affil_dst  = (const int*)d_in[14];
  int nWrites = in_sizes[7];
  int nCites  = in_sizes[9];
  int nRev    = in_sizes[11];
  int nAffil  = in_sizes[13];

  float* ws = (float*)d_ws;
  float* agg_writes = ws + OFF_AGG_WRITES;
  float* agg_cites  = ws + OFF_AGG_CITES;
  float* agg_rev    = ws + OFF_AGG_REV;
  float* agg_affil  = ws + OFF_AGG_AFFIL;
  float* deg_writes = ws + OFF_DEG_WRITES;
  float* deg_cites  = ws + OFF_DEG_CITES;
  float* deg_rev    = ws + OFF_DEG_REV;
  float* deg_affil  = ws + OFF_DEG_AFFIL;
  float* weights    = ws + OFF_WEIGHTS;       // 4 x 128 x 128
  const float* W_affil  = weights + 0 * F * F;
  const float* W_cites  = weights + 1 * F * F;
  const float* W_rev    = weights + 2 * F * F;
  const float* W_writes = weights + 3 * F * F;

  float* out        = (float*)d_out;
  float* out_author = out;
  float* out_inst   = out + SZ_AUTHOR;
  float* out_paper  = out + SZ_AUTHOR + SZ_INST;

  // Phase 0: zero accumulators + degrees
  {
    long n4 = (long)(ZERO_FLOATS / 4);
    int blocks = (int)((n4 + 255) / 256);
    zero_kernel<<<blocks, 256, 0, stream>>>((float4*)ws, n4);
  }
  // Phase 1: basis-combined weights
  make_weights<<<(4 * F * F + 255) / 256, 256, 0, stream>>>(basis_w, w_comp, weights);

  // Phase 2: edge scatter (one wave / edge; 8 edges per 256-thread block)
  edge_scatter<<<(nWrites + 7) / 8, 256, 0, stream>>>(x_author, writes_src, writes_dst,
                                                      agg_writes, deg_writes, nWrites);
  edge_scatter<<<(nCites + 7) / 8, 256, 0, stream>>>(x_paper, cites_src, cites_dst,
                                                     agg_cites, deg_cites, nCites);
  edge_scatter<<<(nRev + 7) / 8, 256, 0, stream>>>(x_paper, rev_src, rev_dst,
                                                   agg_rev, deg_rev, nRev);
  edge_scatter<<<(nAffil + 7) / 8, 256, 0, stream>>>(x_author, affil_src, affil_dst,
                                                     agg_affil, deg_affil, nAffil);

  // Phase 3: WMMA GEMM + bias + relu per dst type (32 rows per wave)
  {
    int strips = N_AUTHOR / 32;               // 9375
    gemm_finalize<<<(strips + 7) / 8, 256, 0, stream>>>(
        strips,
        agg_rev, W_rev, deg_rev,
        x_author, self_w, nullptr,
        nullptr, nullptr, nullptr,
        bias, out_author);
  }
  {
    int strips = N_INST / 32;                 // 250
    gemm_finalize<<<(strips + 7) / 8, 256, 0, stream>>>(
        strips,
        agg_affil, W_affil, deg_affil,
        x_inst, self_w, nullptr,
        nullptr, nullptr, nullptr,
        bias, out_inst);
  }
  {
    int strips = N_PAPER / 32;                // 15625
    gemm_finalize<<<(strips + 7) / 8, 256, 0, stream>>>(
        strips,
        agg_writes, W_writes, deg_writes,
        agg_cites, W_cites, deg_cites,
        x_paper, self_w, nullptr,
        bias, out_paper);
  }
}